// TimeWindowBlock_3925600108827
// MI455X (gfx1250) — compile-verified
//
#include <hip/hip_runtime.h>
#include <stdint.h>

typedef float v2f __attribute__((ext_vector_type(2)));
typedef float v8f __attribute__((ext_vector_type(8)));
typedef int   v4i __attribute__((ext_vector_type(4)));

typedef __attribute__((address_space(1))) v4i* gv4i_ptr;   // global
typedef __attribute__((address_space(3))) v4i* lv4i_ptr;   // LDS

#define H 64
#define FF 36
#define BB 512
#define WW 8
#define SS 200
#define SPAD 208        // 13 * 16 (s padded)
#define OPAD 48         // 3 * 16  (o padded)
#define KSTRIDE 68      // padded row stride (floats); 68 % 64 == 4 -> conflict-free fragment loads
#define NTHREADS 256
#define NWAVES 8
#define NPAIRS 39       // 13 s-tiles * 3 o-tiles

#if defined(__HIP_DEVICE_COMPILE__) && __has_builtin(__builtin_amdgcn_global_load_async_to_lds_b128)
#define HAVE_ASYNC_LDS 1
#else
#define HAVE_ASYNC_LDS 0
#endif

#if defined(__HIP_DEVICE_COMPILE__) && __has_builtin(__builtin_amdgcn_s_wait_asynccnt)
#define WAIT_ASYNC() __builtin_amdgcn_s_wait_asynccnt(0)
#else
#define WAIT_ASYNC() asm volatile("s_wait_asynccnt 0" ::: "memory")
#endif

__global__ __launch_bounds__(NTHREADS)
void twb_kernel(const float* __restrict__ query,
                const float* __restrict__ key,
                const float* __restrict__ W1,
                const float* __restrict__ b1,
                const float* __restrict__ prelu_a,
                const float* __restrict__ W2,
                const float* __restrict__ b2,
                const uint8_t* __restrict__ mask,
                float* __restrict__ out)
{
    extern __shared__ float lds[];
    float* kslab = lds;                        // SPAD * KSTRIDE
    float* Meff  = kslab + SPAD * KSTRIDE;     // OPAD * KSTRIDE
    float* hq    = Meff  + OPAD * KSTRIDE;     // OPAD
    float* w2s   = hq    + OPAD;               // OPAD
    float* qs    = w2s   + OPAD;               // H
    float* part  = qs    + H;                  // 3 * SPAD
    float* wgt   = part  + 3 * SPAD;           // SPAD
    float* sred  = wgt   + SPAD;               // NTHREADS
    float* outp  = sred  + NTHREADS;           // 4 * H

    const int tid = threadIdx.x;
    const int bw  = blockIdx.x;                // b*W + w
    const int b   = bw / WW;

    const float* kbase = key + (size_t)bw * (SS * H);

    // ---- q[b] into LDS
    if (tid < H) qs[tid] = query[b * H + tid];

    // ---- key tile HBM -> LDS (read exactly once); async path overlaps with setup math
#if HAVE_ASYNC_LDS
    for (int i = tid; i < SS * 16; i += NTHREADS) {
        const int s = i >> 4, c = i & 15;      // 16 x b128 per row
        __builtin_amdgcn_global_load_async_to_lds_b128(
            (gv4i_ptr)(kbase + s * H + c * 4),
            (lv4i_ptr)(kslab + s * KSTRIDE + c * 4),
            0, 0);
    }
#else
    {
        const float4* src = (const float4*)kbase;
        for (int i = tid; i < SS * 16; i += NTHREADS) {
            const int s = i >> 4, c = i & 15;
            *(float4*)(kslab + s * KSTRIDE + c * 4) = src[i];
        }
    }
#endif

    __syncthreads();   // qs visible (async key-tile still in flight)

    // ---- Meff[o][j] = W1b - W1c + W1d * q_j   (zero rows for o >= FF)
    for (int i = tid; i < OPAD * H; i += NTHREADS) {
        const int o = i >> 6, j = i & (H - 1);
        float v = 0.0f;
        if (o < FF) {
            const float* row = W1 + o * (4 * H);
            v = row[H + j] - row[2 * H + j] + row[3 * H + j] * qs[j];
        }
        Meff[o * KSTRIDE + j] = v;
    }
    // ---- hq[o] = b1[o] + sum_j (W1a + W1c) * q_j ; W2 padded with zeros
    for (int o = tid; o < OPAD; o += NTHREADS) {
        float acc = 0.0f;
        if (o < FF) {
            const float* row = W1 + o * (4 * H);
            acc = b1[o];
            for (int j = 0; j < H; ++j)
                acc += (row[j] + row[2 * H + j]) * qs[j];
        }
        hq[o]  = acc;
        w2s[o] = (o < FF) ? W2[o] : 0.0f;
    }

#if HAVE_ASYNC_LDS
    WAIT_ASYNC();      // drain this wave's async tile loads
#endif
    __syncthreads();   // kslab + Meff/hq/w2 visible to all waves

    const float pa   = prelu_a[0];
    const int   wave = tid >> 5;
    const int   lane = tid & 31;
    const int   l    = lane & 15;
    const int   half = lane >> 4;

    // ---- WMMA GEMM: h[s,o] = hq[o] + sum_j kslab[s][j] * Meff[o][j]
    //      A = k-tile (16x4 f32), B = Meff^T slice (4x16 f32), C = h (16x16 f32)
    for (int p = wave; p < NPAIRS; p += NWAVES) {
        const int t = p / 3, u = p % 3;
        const int s0 = t * 16, o0 = u * 16;

        v8f c;
        const float hv = hq[o0 + l];
#pragma unroll
        for (int r = 0; r < 8; ++r) c[r] = hv;

#pragma unroll
        for (int kk = 0; kk < 16; ++kk) {
            const int j0 = kk * 4 + 2 * half;  // lanes 0-15: K=0,1  lanes 16-31: K=2,3
            v2f a, bm;
            a.x  = kslab[(s0 + l) * KSTRIDE + j0];
            a.y  = kslab[(s0 + l) * KSTRIDE + j0 + 1];
            bm.x = Meff[(o0 + l) * KSTRIDE + j0];
            bm.y = Meff[(o0 + l) * KSTRIDE + j0 + 1];
            c = __builtin_amdgcn_wmma_f32_16x16x4_f32(
                    false, a, false, bm, (short)0, c, false, false);
        }

        // score partial: sum_o w2[o] * prelu(h[s,o]) over this o-tile
        const float w2v = w2s[o0 + l];
#pragma unroll
        for (int r = 0; r < 8; ++r) {
            float h = c[r];
            h = (h >= 0.0f) ? h : pa * h;
            float v = w2v * h;
            v += __shfl_xor(v, 8, 16);
            v += __shfl_xor(v, 4, 16);
            v += __shfl_xor(v, 2, 16);
            v += __shfl_xor(v, 1, 16);
            if (l == 0)
                part[u * SPAD + (s0 + r + 8 * half)] = v;   // each (u,s) written once
        }
    }
    __syncthreads();

    // ---- masked softmax over s
    const uint8_t* mrow = mask + (size_t)bw * SS;
    float myval = -1e30f;
    if (tid < SS) {
        float sc = b2[0] + part[0 * SPAD + tid] + part[1 * SPAD + tid] + part[2 * SPAD + tid];
        if (mrow[tid]) sc = -10000.0f;
        myval = sc;
    }
    sred[tid] = myval;
    __syncthreads();
    for (int st = NTHREADS / 2; st > 0; st >>= 1) {
        if (tid < st) sred[tid] = fmaxf(sred[tid], sred[tid + st]);
        __syncthreads();
    }
    const float mx = sred[0];
    __syncthreads();
    const float e = (tid < SS) ? __expf(myval - mx) : 0.0f;
    sred[tid] = e;
    __syncthreads();
    for (int st = NTHREADS / 2; st > 0; st >>= 1) {
        if (tid < st) sred[tid] += sred[tid + st];
        __syncthreads();
    }
    const float inv = 1.0f / sred[0];
    if (tid < SPAD) wgt[tid] = (tid < SS) ? e * inv : 0.0f;
    __syncthreads();

    // ---- out[b,w,:] = sum_s wgt[s] * kslab[s][:]  (key reused from LDS)
    {
        const int j  = tid & (H - 1);
        const int ch = tid >> 6;               // 0..3, 50 s-values each
        float acc = 0.0f;
        for (int s = ch * 50; s < ch * 50 + 50; ++s)
            acc += wgt[s] * kslab[s * KSTRIDE + j];
        outp[ch * H + j] = acc;
    }
    __syncthreads();
    if (tid < H)
        out[(size_t)bw * H + tid] =
            outp[tid] + outp[H + tid] + outp[2 * H + tid] + outp[3 * H + tid];
}

extern "C" void kernel_launch(void* const* d_in, const int* in_sizes, int n_in,
                              void* d_out, int out_size, void* d_ws, size_t ws_size,
                              hipStream_t stream) {
    (void)in_sizes; (void)n_in; (void)out_size; (void)d_ws; (void)ws_size;
    const float*   query = (const float*)d_in[0];
    const float*   key   = (const float*)d_in[1];
    const float*   W1    = (const float*)d_in[2];
    const float*   b1    = (const float*)d_in[3];
    const float*   pa    = (const float*)d_in[4];
    const float*   W2    = (const float*)d_in[5];
    const float*   b2    = (const float*)d_in[6];
    const uint8_t* mask  = (const uint8_t*)d_in[7];
    float*         out   = (float*)d_out;

    const size_t lds_bytes =
        (size_t)(SPAD * KSTRIDE + OPAD * KSTRIDE + OPAD + OPAD + H + 3 * SPAD + SPAD
                 + NTHREADS + 4 * H) * sizeof(float);   // ~76 KB, 4 blocks / 320 KB WGP

    hipLaunchKernelGGL(twb_kernel, dim3(BB * WW), dim3(NTHREADS), lds_bytes, stream,
                       query, key, W1, b1, pa, W2, b2, mask, out);
}